// InterGraphAttention_4501125726641
// MI455X (gfx1250) — compile-verified
//
#include <hip/hip_runtime.h>
#include <hip/hip_bf16.h>
#include <math.h>

typedef __attribute__((ext_vector_type(2))) float v2f;
typedef __attribute__((ext_vector_type(8))) float v8f;

#define DD 256      // feature dim
#define LDA 260     // padded LDS row stride (floats): bank-conflict-free
#define ROWS 32     // rows per GEMM block (2 row-tiles), LDS = 32*260*4 = 33,280 B
#define NEG_BIG (-3.0e38f)

// ---------------------------------------------------------------- helpers
__device__ __forceinline__ float eluf(float x) {
    return x > 0.f ? x : (expf(x) - 1.f);
}
__device__ __forceinline__ float lrelu02(float x) {
    return x > 0.f ? x : 0.2f * x;
}
// order-preserving float atomic max using integer atomics.
// Branch on the sign BIT so -0.0 takes the unsigned-min path (v>=0 would map
// -0.0 to INT_MIN and never win an atomicMax).
__device__ __forceinline__ void atomicMaxF(float* addr, float v) {
    int bits = __float_as_int(v);
    if (bits >= 0) atomicMax((int*)addr, bits);
    else           atomicMin((unsigned int*)addr, (unsigned int)bits);
}

// ---------------------------------------------------------------- init
// d_out -> 0 ; mz region: m1(2N)=-BIG, z1(2N)=0, m2(2N)=-BIG, z2(2N)=0, S(2)=0
__global__ void init_kernel(float* __restrict__ out, float* __restrict__ mz,
                            long long outLen, int Nn) {
    long long stride = (long long)gridDim.x * blockDim.x;
    long long total  = outLen + 8LL * Nn + 2;
    for (long long i = (long long)blockIdx.x * blockDim.x + threadIdx.x;
         i < total; i += stride) {
        if (i < outLen) {
            out[i] = 0.f;
        } else {
            long long k = i - outLen;
            float v = 0.f;
            if (k < 2LL * Nn)                       v = NEG_BIG;   // m1
            else if (k >= 4LL * Nn && k < 6LL * Nn) v = NEG_BIG;   // m2
            mz[k] = v;
        }
    }
}

// ---------------------------------------------------------------- fused ELU + GEMM (WMMA f32)
// Y[n, 0:64]  = elu(X[n,:]) @ W_src      (hs projection)
// Y[n, 64:128]= elu(X[n,:]) @ W_dst      (hd projection)
// one block = 32 output rows (2 row-tiles in LDS); 8 waves, each wave owns
// (row-tile = wv>>2, column tiles {2c,2c+1} with c = wv&3) -> 2 accumulators.
__global__ __launch_bounds__(256) void gemm_elu_wmma(
        const float* __restrict__ X,
        const float* __restrict__ Wsrc, const float* __restrict__ Wdst,
        float* __restrict__ Y, int Nn) {
    __shared__ float Als[ROWS * LDA];
    const int rowBase = blockIdx.x * ROWS;

    // stage 32x256 A tile with ELU applied (coalesced global reads)
    for (int i = threadIdx.x; i < ROWS * DD; i += 256) {
        int r = i >> 8, c = i & 255;
        int rr = rowBase + r; if (rr >= Nn) rr = Nn - 1;
        Als[r * LDA + c] = eluf(X[(size_t)rr * DD + c]);
    }
    __syncthreads();

    const int lane  = threadIdx.x & 31;
    const int wv    = threadIdx.x >> 5;
    const int rt    = wv >> 2;          // row tile 0/1
    const int cpair = wv & 3;           // column tiles {2*cpair, 2*cpair+1}
    const int nloc  = lane & 15;        // M/N index within 16-tile
    const int hi    = lane >> 4;        // 0: K{0,1}, 1: K{2,3}

    const float* As = &Als[(rt * 16 + nloc) * LDA];

    // cols 32*cpair .. 32*cpair+31 ; cpair 0,1 -> W_src half, 2,3 -> W_dst half
    const float* Wb = (cpair < 2) ? Wsrc : Wdst;
    const int ccl0  = (cpair & 1) * 32 + nloc;   // column within the 64-wide half
    const int ccl1  = ccl0 + 16;

    v8f acc0 = {0.f, 0.f, 0.f, 0.f, 0.f, 0.f, 0.f, 0.f};
    v8f acc1 = {0.f, 0.f, 0.f, 0.f, 0.f, 0.f, 0.f, 0.f};
    #pragma unroll 4
    for (int k = 0; k < DD; k += 4) {
        const int k0 = k + hi * 2;
        v2f a, b0, b1;
        // A 16x4 (32-bit): lane row M=nloc, VGPR0=K0, VGPR1=K0+1
        a.x = As[k0];
        a.y = As[k0 + 1];
        // B 4x16 (32-bit): row K striped across lanes within a VGPR
        b0.x = Wb[(size_t)k0 * 64 + ccl0];
        b0.y = Wb[(size_t)(k0 + 1) * 64 + ccl0];
        b1.x = Wb[(size_t)k0 * 64 + ccl1];
        b1.y = Wb[(size_t)(k0 + 1) * 64 + ccl1];
        acc0 = __builtin_amdgcn_wmma_f32_16x16x4_f32(
                   false, a, false, b0, (short)0, acc0, false, false);
        acc1 = __builtin_amdgcn_wmma_f32_16x16x4_f32(
                   false, a, false, b1, (short)0, acc1, false, false);
    }

    // C/D 16x16 f32: VGPR v -> M=v (lanes 0-15), M=v+8 (lanes 16-31); N=lane%16
    const int rOff    = rt * 16 + hi * 8;
    const int colBase = cpair * 32 + nloc;
    if (rowBase + ROWS <= Nn) {
        // full tile fast path: clean store clause, no per-row exec masking
        float* Yp = Y + (size_t)(rowBase + rOff) * 128 + colBase;
        #pragma unroll
        for (int v = 0; v < 8; ++v) {
            Yp[(size_t)v * 128]      = acc0[v];
            Yp[(size_t)v * 128 + 16] = acc1[v];
        }
    } else {
        #pragma unroll
        for (int v = 0; v < 8; ++v) {
            int row = rowBase + rOff + v;
            if (row < Nn) {
                Y[(size_t)row * 128 + colBase]      = acc0[v];
                Y[(size_t)row * 128 + colBase + 16] = acc1[v];
            }
        }
    }
}

// ---------------------------------------------------------------- per-node attention dots
__global__ void avec_kernel(const float* __restrict__ Y,
                            const float* __restrict__ attS,
                            const float* __restrict__ attD,
                            float* __restrict__ aS, float* __restrict__ aD, int Nn) {
    int i = blockIdx.x * blockDim.x + threadIdx.x;
    if (i >= Nn * 2) return;
    int node = i >> 1, h = i & 1;
    const float* y = Y + (size_t)node * 128;
    float s = 0.f, d = 0.f;
    #pragma unroll 8
    for (int c = 0; c < 32; ++c) {
        s += y[h * 32 + c]      * attS[h * 32 + c];
        d += y[64 + h * 32 + c] * attD[h * 32 + c];
    }
    aS[i] = s; aD[i] = d;
}

// ---------------------------------------------------------------- edge pass 1: segment max
__global__ void edge_max(const int* __restrict__ es, const int* __restrict__ ed,
                         const float* __restrict__ aS, const float* __restrict__ aD,
                         float* __restrict__ m, int E, int Nn) {
    int e = blockIdx.x * blockDim.x + threadIdx.x;
    if (e >= E + Nn) return;
    int s, d;
    if (e < E) { s = es[e]; d = ed[e]; } else { s = d = e - E; }
    #pragma unroll
    for (int h = 0; h < 2; ++h) {
        float a = lrelu02(aS[s * 2 + h] + aD[d * 2 + h]);
        atomicMaxF(&m[d * 2 + h], a);
    }
}

// ---------------------------------------------------------------- edge pass 2: sum of exp
__global__ void edge_sum(const int* __restrict__ es, const int* __restrict__ ed,
                         const float* __restrict__ aS, const float* __restrict__ aD,
                         const float* __restrict__ m, float* __restrict__ z,
                         int E, int Nn) {
    int e = blockIdx.x * blockDim.x + threadIdx.x;
    if (e >= E + Nn) return;
    int s, d;
    if (e < E) { s = es[e]; d = ed[e]; } else { s = d = e - E; }
    #pragma unroll
    for (int h = 0; h < 2; ++h) {
        float a = lrelu02(aS[s * 2 + h] + aD[d * 2 + h]);
        atomicAdd(&z[d * 2 + h], expf(a - m[d * 2 + h]));
    }
}

// ---------------------------------------------------------------- edge pass 3: weighted scatter
// one wave32 per edge: lane -> channel (head0), lane+32 -> channel (head1)
__global__ __launch_bounds__(256) void edge_scatter(
        const int* __restrict__ es, const int* __restrict__ ed,
        const float* __restrict__ aS, const float* __restrict__ aD,
        const float* __restrict__ m, const float* __restrict__ z,
        const float* __restrict__ Ysrc, float* __restrict__ out, int E, int Nn) {
    int wid  = blockIdx.x * (blockDim.x >> 5) + (threadIdx.x >> 5);
    int lane = threadIdx.x & 31;
    if (wid >= E + Nn) return;
    int s, d;
    if (wid < E) { s = es[wid]; d = ed[wid]; } else { s = d = wid - E; }
    float c0, c1;
    {
        float a0 = lrelu02(aS[s * 2 + 0] + aD[d * 2 + 0]);
        float a1 = lrelu02(aS[s * 2 + 1] + aD[d * 2 + 1]);
        c0 = expf(a0 - m[d * 2 + 0]) / (z[d * 2 + 0] + 1e-16f);
        c1 = expf(a1 - m[d * 2 + 1]) / (z[d * 2 + 1] + 1e-16f);
    }
    const float* ys = Ysrc + (size_t)s * 128;   // hs = first 64 columns
    float* o = out + (size_t)d * 64;
    atomicAdd(&o[lane],      ys[lane]      * c0);
    atomicAdd(&o[lane + 32], ys[lane + 32] * c1);
}

// ---------------------------------------------------------------- enrichment scalar reduction
__global__ void enrich_reduce(const float* __restrict__ ew, int E, float* __restrict__ S) {
    __shared__ float s1[256], s2[256];
    float l1 = 0.f, l2 = 0.f;
    for (int i = blockIdx.x * blockDim.x + threadIdx.x; i < E;
         i += gridDim.x * blockDim.x) {
        float w = fminf(fmaxf(ew[i], 0.3f), 3.0f);
        float e = expf(w);
        l1 += e; l2 += e * w;
    }
    s1[threadIdx.x] = l1; s2[threadIdx.x] = l2;
    __syncthreads();
    for (int o = 128; o > 0; o >>= 1) {
        if (threadIdx.x < o) {
            s1[threadIdx.x] += s1[threadIdx.x + o];
            s2[threadIdx.x] += s2[threadIdx.x + o];
        }
        __syncthreads();
    }
    if (threadIdx.x == 0) { atomicAdd(&S[0], s1[0]); atomicAdd(&S[1], s2[0]); }
}

// ---------------------------------------------------------------- final: (+bias) * factor
__global__ void final_scale(float* __restrict__ out, const float* __restrict__ bias,
                            const float* __restrict__ S, const float* __restrict__ esc,
                            long long total) {
    float factor = 1.f + 0.5f * tanhf(esc[0]) * (S[1] / S[0] - 1.f);
    long long stride = (long long)gridDim.x * blockDim.x;
    for (long long i = (long long)blockIdx.x * blockDim.x + threadIdx.x;
         i < total; i += stride) {
        out[i] = (out[i] + bias[i & 63]) * factor;
    }
}

// ---------------------------------------------------------------- launcher
extern "C" void kernel_launch(void* const* d_in, const int* in_sizes, int n_in,
                              void* d_out, int out_size, void* d_ws, size_t ws_size,
                              hipStream_t stream) {
    const float* h_x  = (const float*)d_in[0];
    const float* t_x  = (const float*)d_in[1];
    const int*   ei   = (const int*)  d_in[2];
    const float* ew   = (const float*)d_in[3];
    const float* Wsrc = (const float*)d_in[4];
    const float* Wdst = (const float*)d_in[5];
    const float* attS = (const float*)d_in[6];
    const float* attD = (const float*)d_in[7];
    const float* bias = (const float*)d_in[8];
    const float* esc  = (const float*)d_in[9];

    const int Nn = in_sizes[0] / DD;      // 100000
    const int E  = in_sizes[3];           // 1600000
    const int* srcA = ei;
    const int* dstA = ei + E;

    float* out_h = (float*)d_out;                    // h_rep (direction 2)
    float* out_t = (float*)d_out + (size_t)Nn * 64;  // t_rep (direction 1)

    // workspace layout (floats)
    float* w   = (float*)d_ws;
    float* Yh  = w;                                // N x 128
    float* Yt  = Yh  + (size_t)Nn * 128;           // N x 128
    float* aSh = Yt  + (size_t)Nn * 128;           // N x 2
    float* aDh = aSh + (size_t)Nn * 2;
    float* aSt = aDh + (size_t)Nn * 2;
    float* aDt = aSt + (size_t)Nn * 2;
    float* mz  = aDt + (size_t)Nn * 2;             // m1,z1,m2,z2,S
    float* m1  = mz;
    float* z1  = m1 + (size_t)Nn * 2;
    float* m2  = z1 + (size_t)Nn * 2;
    float* z2  = m2 + (size_t)Nn * 2;
    float* S   = z2 + (size_t)Nn * 2;

    const long long outLen = 2LL * Nn * 64;

    // 1) init accumulators
    init_kernel<<<4096, 256, 0, stream>>>((float*)d_out, mz, outLen, Nn);

    // 2) fused ELU + projection GEMMs (WMMA f32)
    const int gB = (Nn + ROWS - 1) / ROWS;
    gemm_elu_wmma<<<gB, 256, 0, stream>>>(h_x, Wsrc, Wdst, Yh, Nn);
    gemm_elu_wmma<<<gB, 256, 0, stream>>>(t_x, Wsrc, Wdst, Yt, Nn);

    // 3) per-node attention scalars
    const int aB = (Nn * 2 + 255) / 256;
    avec_kernel<<<aB, 256, 0, stream>>>(Yh, attS, attD, aSh, aDh, Nn);
    avec_kernel<<<aB, 256, 0, stream>>>(Yt, attS, attD, aSt, aDt, Nn);

    // 4) edge passes
    const int Et  = E + Nn;
    const int eB  = (Et + 255) / 256;        // thread per edge
    const int eWB = (Et + 7) / 8;            // wave per edge (8 waves / block)

    // direction 1: h -> t  (src=srcA, dst=dstA), output t_rep
    edge_max    <<<eB,  256, 0, stream>>>(srcA, dstA, aSh, aDt, m1, E, Nn);
    edge_sum    <<<eB,  256, 0, stream>>>(srcA, dstA, aSh, aDt, m1, z1, E, Nn);
    edge_scatter<<<eWB, 256, 0, stream>>>(srcA, dstA, aSh, aDt, m1, z1, Yh, out_t, E, Nn);

    // direction 2: t -> h  (reversed edges), output h_rep
    edge_max    <<<eB,  256, 0, stream>>>(dstA, srcA, aSt, aDh, m2, E, Nn);
    edge_sum    <<<eB,  256, 0, stream>>>(dstA, srcA, aSt, aDh, m2, z2, E, Nn);
    edge_scatter<<<eWB, 256, 0, stream>>>(dstA, srcA, aSt, aDh, m2, z2, Yt, out_h, E, Nn);

    // 5) enrichment scalar + final scale/bias
    enrich_reduce<<<1024, 256, 0, stream>>>(ew, E, S);
    final_scale<<<4096, 256, 0, stream>>>((float*)d_out, bias, S, esc, outLen);
}